// GATClassifier_53111565582469
// MI455X (gfx1250) — compile-verified
//
#include <hip/hip_runtime.h>
#include <hip/hip_bf16.h>

typedef float v2f __attribute__((ext_vector_type(2)));
typedef float v8f __attribute__((ext_vector_type(8)));
typedef unsigned int u32x4 __attribute__((ext_vector_type(4)));
typedef int i32x4 __attribute__((ext_vector_type(4)));
typedef int i32x8 __attribute__((ext_vector_type(8)));

#define NEG_SLOPE 0.2f
#define EPS 1e-16f
#define LDSPITCH 132   // 128 + 4 pad DWORDs inserted by TDM -> conflict-free LDS

// ---------------------------------------------------------------------------
// Utility kernels
// ---------------------------------------------------------------------------
__global__ void zero_f32(float* p, int n) {
  int i = blockIdx.x * blockDim.x + threadIdx.x;
  if (i < n) p[i] = 0.0f;
}

__global__ void bias_init(float* __restrict__ out, const float* __restrict__ b,
                          int N, int HC) {
  int i = blockIdx.x * blockDim.x + threadIdx.x;
  if (i < N * HC) out[i] = b[i % HC];
}

__global__ void relu_k(float* p, int n) {
  int i = blockIdx.x * blockDim.x + threadIdx.x;
  if (i < n) p[i] = fmaxf(p[i], 0.0f);
}

// edge_attr column sums (for self-loop fill value = mean)
__global__ void ea_sum(const float* __restrict__ ea, float* sums, int E) {
  float s0 = 0.f, s1 = 0.f, s2 = 0.f;
  for (int e = blockIdx.x * blockDim.x + threadIdx.x; e < E;
       e += gridDim.x * blockDim.x) {
    s0 += ea[e * 3 + 0];
    s1 += ea[e * 3 + 1];
    s2 += ea[e * 3 + 2];
  }
  atomicAdd(&sums[0], s0);
  atomicAdd(&sums[1], s1);
  atomicAdd(&sums[2], s2);
}

// Q[h][k] = sum_c We[(h*C+c)*3 + k] * att_e[h*C+c]   (collapses edge GEMM)
__global__ void compute_q(const float* __restrict__ We,
                          const float* __restrict__ att_e, float* Q, int H, int C) {
  int t = threadIdx.x;
  if (t >= H * 3) return;
  int hd = t / 3, k = t - hd * 3;
  float q = 0.f;
  for (int c = 0; c < C; ++c) q += We[(hd * C + c) * 3 + k] * att_e[hd * C + c];
  Q[t] = q;
}

// ---------------------------------------------------------------------------
// Node transform: Out[N, Mcols] = X[N,128] @ W[Mcols,128]^T  via f32 WMMA.
// Per block: TDM stages the shared 16x128 A tile into padded LDS once;
// each of the 4 waves computes one 16x16 output tile (K=128 -> 32 WMMAs).
// ---------------------------------------------------------------------------
__global__ void gemm_wmma_k128(const float* __restrict__ X,
                               const float* __restrict__ W,
                               float* __restrict__ Out,
                               int Nrows, int Mcols, int ldOut) {
  __shared__ float shA[16 * LDSPITCH];

  const int wave   = (int)(threadIdx.x >> 5);
  const int lane   = (int)(threadIdx.x & 31);
  const int nWaves = (int)(blockDim.x >> 5);
  const int row0   = (int)blockIdx.x * 16;
  if (row0 >= Nrows) return;   // block-uniform

  // ---- TDM: async DMA of A tile (16 rows x 128 f32) global -> LDS --------
  if (wave == 0) {
    unsigned lds = (unsigned)(size_t)(&shA[0]);
    unsigned long long ga =
        (unsigned long long)(size_t)(X + (size_t)row0 * 128);
    int rowsLeft = Nrows - row0;

    u32x4 g0;
    g0.x = 1u;                                     // count=1 valid descriptor
    g0.y = lds;                                    // lds_addr (bytes)
    g0.z = (unsigned)(ga & 0xFFFFFFFFu);           // global_addr[31:0]
    g0.w = (unsigned)((ga >> 32) & 0x01FFFFFFu)    // global_addr[56:32]
           | (2u << 30);                           // type = 2 ("image")

    i32x8 g1;
    g1[0] = (2 << 16)        // data_size = 4 bytes
          | (1 << 20)        // pad_enable
          | (6 << 22)        // pad_interval: every 128 DWORDs
          | (3 << 25);       // pad_amount: 4 DWORDs  -> LDS pitch 132 floats
    g1[1] = (128 & 0xFFFF) << 16;                          // tensor_dim0 = 128
    g1[2] = (int)(((unsigned)rowsLeft & 0xFFFFu) << 16);   // tensor_dim1 lo16
    g1[3] = (int)((((unsigned)rowsLeft >> 16) & 0xFFFFu)   // tensor_dim1 hi16
           | (128u << 16));                                // tile_dim0 = 128
    g1[4] = 16;                                            // tile_dim1=16, tile_dim2=0
    g1[5] = 128;                                           // tensor_dim0_stride lo32
    g1[6] = 0;
    g1[7] = 0;
    i32x4 gz4 = {0, 0, 0, 0};
    i32x8 gz8 = {0, 0, 0, 0, 0, 0, 0, 0};
    __builtin_amdgcn_tensor_load_to_lds(g0, g1, gz4, gz4, gz8, 0);
    __builtin_amdgcn_s_wait_tensorcnt(0);
  }
  __syncthreads();

  const int col0 = ((int)blockIdx.y * nWaves + wave) * 16;
  if (col0 >= ((Mcols + 15) & ~15)) return;   // wave-uniform exit (after barrier)

  const int half = lane >> 4;      // which K-pair / which output row-half
  const int r    = lane & 15;

  // Invalid B rows (layer 3, Mcols=10): clamp pointer to row 0. The garbage
  // only lands in output column n=r which the store guard below discards.
  const int brow = (col0 + r < Mcols) ? (col0 + r) : 0;

  const float* ap = &shA[r * LDSPITCH + 2 * half];
  const float* wp = W + (size_t)brow * 128 + 2 * half;

  v8f acc = {};
#pragma unroll
  for (int k = 0; k < 128; k += 4) {
    v2f av, bv;
    av.x = ap[k];
    av.y = ap[k + 1];
    bv.x = wp[k];
    bv.y = wp[k + 1];
    acc = __builtin_amdgcn_wmma_f32_16x16x4_f32(false, av, false, bv,
                                                (short)0, acc, false, false);
  }

  const int col = col0 + r;
  if (col < Mcols) {
#pragma unroll
    for (int j = 0; j < 8; ++j) {
      int rowm = row0 + j + 8 * half;            // C/D: VGPR j holds rows j, j+8
      if (rowm < Nrows) Out[(size_t)rowm * ldOut + col] = acc[j];
    }
  }
}

// ---------------------------------------------------------------------------
// Per-node attention projections: as/ad[n,h] = <h[n,h,:], att_{s,d}[h,:]>
// ---------------------------------------------------------------------------
__global__ void node_alpha(const float* __restrict__ h,
                           const float* __restrict__ att_s,
                           const float* __restrict__ att_d,
                           float* __restrict__ as_, float* __restrict__ ad_,
                           int N, int H, int C) {
  int t = blockIdx.x * blockDim.x + threadIdx.x;
  if (t >= N * H) return;
  int n = t / H, hd = t - n * H;
  const float* hp = h + (size_t)n * (H * C) + hd * C;
  const float* sp = att_s + hd * C;
  const float* dp = att_d + hd * C;
  float a = 0.f, b = 0.f;
  for (int c = 0; c < C; ++c) {
    float v = hp[c];
    a += v * sp[c];
    b += v * dp[c];
  }
  as_[t] = a;
  ad_[t] = b;
}

__device__ __forceinline__ unsigned fkey_enc(float f) {
  unsigned u = __float_as_uint(f);
  return (u & 0x80000000u) ? ~u : (u | 0x80000000u);   // monotonic total order
}
__device__ __forceinline__ float fkey_dec(unsigned u) {
  return __uint_as_float((u & 0x80000000u) ? (u & 0x7FFFFFFFu) : ~u);
}

// Pass 1: thread-per-edge; all H logits, leaky-relu, atomic segment max
template <int H>
__global__ void edge_logits_t(const int* __restrict__ srcA, const int* __restrict__ dstA,
                              const float* __restrict__ ea,
                              const float* __restrict__ easum, float invE,
                              const float* __restrict__ Q,
                              const float* __restrict__ as_, const float* __restrict__ ad_,
                              float* __restrict__ a_edge, unsigned* __restrict__ mkey,
                              int E, int N) {
  int e = blockIdx.x * blockDim.x + threadIdx.x;
  if (e >= E + N) return;
  int s, d;
  float e0, e1, e2;
  if (e < E) {
    s = srcA[e]; d = dstA[e];
    e0 = ea[e * 3 + 0]; e1 = ea[e * 3 + 1]; e2 = ea[e * 3 + 2];
  } else {                      // synthesized self-loop, edge_attr = mean
    s = d = e - E;
    e0 = easum[0] * invE; e1 = easum[1] * invE; e2 = easum[2] * invE;
  }
  const float* sp = as_ + (size_t)s * H;
  const float* dp = ad_ + (size_t)d * H;
#pragma unroll
  for (int hd = 0; hd < H; ++hd) {
    float a = sp[hd] + dp[hd] +
              e0 * Q[hd * 3 + 0] + e1 * Q[hd * 3 + 1] + e2 * Q[hd * 3 + 2];
    a = (a > 0.f) ? a : NEG_SLOPE * a;
    a_edge[(size_t)e * H + hd] = a;
    atomicMax(&mkey[d * H + hd], fkey_enc(a));
  }
}

// Pass 2: ex = exp(a - m[dst]); accumulate denominator
template <int H>
__global__ void edge_exp_t(const int* __restrict__ dstA,
                           float* __restrict__ a_edge,
                           const unsigned* __restrict__ mkey,
                           float* __restrict__ denom, int E, int N) {
  int e = blockIdx.x * blockDim.x + threadIdx.x;
  if (e >= E + N) return;
  int d = (e < E) ? dstA[e] : (e - E);
#pragma unroll
  for (int hd = 0; hd < H; ++hd) {
    float m = fkey_dec(mkey[d * H + hd]);
    float ex = __expf(a_edge[(size_t)e * H + hd] - m);
    a_edge[(size_t)e * H + hd] = ex;
    atomicAdd(&denom[d * H + hd], ex);
  }
}

// Pass 3: out[dst, c] += h[src, c] * alpha. One thread per (edge, 4-channel
// group): float4 gather + 4 f32 global atomics (all L2-resident tables).
template <int H, int C>
__global__ void edge_aggregate_t(const int* __restrict__ srcA, const int* __restrict__ dstA,
                                 const float* __restrict__ hfeat,
                                 const float* __restrict__ a_edge,
                                 const float* __restrict__ denom,
                                 float* __restrict__ out, int E, int N) {
  constexpr int HC = H * C;
  constexpr int G  = (HC + 3) / 4;
  int tid = blockIdx.x * blockDim.x + threadIdx.x;
  if (tid >= (E + N) * G) return;
  int e = tid / G, g = tid - e * G;
  int s, d;
  if (e < E) { s = srcA[e]; d = dstA[e]; } else { s = d = e - E; }
  const float* hp = hfeat + (size_t)s * HC;
  float*       op = out + (size_t)d * HC;
  const int c0 = g * 4;
  if constexpr ((C % 4) == 0) {
    const int hd = c0 / C;                      // group never crosses a head
    float alpha = a_edge[(size_t)e * H + hd] / (denom[d * H + hd] + EPS);
    float4 hv = *(const float4*)(hp + c0);
    atomicAdd(&op[c0 + 0], hv.x * alpha);
    atomicAdd(&op[c0 + 1], hv.y * alpha);
    atomicAdd(&op[c0 + 2], hv.z * alpha);
    atomicAdd(&op[c0 + 3], hv.w * alpha);
  } else {
#pragma unroll
    for (int j = 0; j < 4; ++j) {
      int c = c0 + j;
      if (c < HC) {
        int hd = c / C;
        float alpha = a_edge[(size_t)e * H + hd] / (denom[d * H + hd] + EPS);
        atomicAdd(&op[c], hp[c] * alpha);
      }
    }
  }
}

// ---------------------------------------------------------------------------
// Global mean pool + log_softmax
// ---------------------------------------------------------------------------
__global__ void pool_kernel(const float* __restrict__ h3, const int* __restrict__ batch,
                            float* __restrict__ pooled, float* __restrict__ cnt,
                            int N, int K) {
  int tid = blockIdx.x * blockDim.x + threadIdx.x;
  if (tid >= N * K) return;
  int n = tid / K, c = tid - n * K;
  int g = batch[n];
  atomicAdd(&pooled[g * K + c], h3[tid]);
  if (c == 0) atomicAdd(&cnt[g], 1.0f);
}

__global__ void finalize_k(const float* __restrict__ pooled, const float* __restrict__ cnt,
                           float* __restrict__ out, int NG, int K) {
  int g = blockIdx.x * blockDim.x + threadIdx.x;
  if (g >= NG) return;
  float inv = 1.0f / fmaxf(cnt[g], 1.0f);
  float v[16];
  float mx = -3.0e38f;
  for (int c = 0; c < K; ++c) {
    v[c] = pooled[g * K + c] * inv;
    mx = fmaxf(mx, v[c]);
  }
  float se = 0.f;
  for (int c = 0; c < K; ++c) se += __expf(v[c] - mx);
  float lse = mx + logf(se);
  for (int c = 0; c < K; ++c) out[g * K + c] = v[c] - lse;
}

// ---------------------------------------------------------------------------
// Host orchestration
// ---------------------------------------------------------------------------
static inline unsigned cdiv(long long a, long long b) {
  return (unsigned)((a + b - 1) / b);
}

template <int H, int C>
static void run_layer(const float* in, const float* Wm, const float* Wem,
                      const float* atts, const float* attd, const float* atte,
                      const float* bb, float* hbuf, float* obuf,
                      const int* src, const int* dst, const float* ea,
                      const float* easum, float invE, float* Qbuf,
                      float* as_, float* ad_, float* denom, unsigned* mkey,
                      float* a_edge, int N, int E, bool do_relu,
                      hipStream_t stream) {
  constexpr int HC = H * C;
  constexpr int G  = (HC + 3) / 4;
  const int Et = E + N;
  const int colTiles = (HC + 15) / 16;
  dim3 g(cdiv(N, 16), cdiv(colTiles, 4));
  gemm_wmma_k128<<<g, 128, 0, stream>>>(in, Wm, hbuf, N, HC, HC);
  node_alpha<<<cdiv((long long)N * H, 256), 256, 0, stream>>>(
      hbuf, atts, attd, as_, ad_, N, H, C);
  compute_q<<<1, 32, 0, stream>>>(Wem, atte, Qbuf, H, C);
  zero_f32<<<cdiv((long long)N * H, 256), 256, 0, stream>>>(denom, N * H);
  zero_f32<<<cdiv((long long)N * H, 256), 256, 0, stream>>>((float*)mkey, N * H);
  edge_logits_t<H><<<cdiv(Et, 256), 256, 0, stream>>>(
      src, dst, ea, easum, invE, Qbuf, as_, ad_, a_edge, mkey, E, N);
  edge_exp_t<H><<<cdiv(Et, 256), 256, 0, stream>>>(dst, a_edge, mkey, denom, E, N);
  bias_init<<<cdiv((long long)N * HC, 256), 256, 0, stream>>>(obuf, bb, N, HC);
  edge_aggregate_t<H, C><<<cdiv((long long)Et * G, 256), 256, 0, stream>>>(
      src, dst, hbuf, a_edge, denom, obuf, E, N);
  if (do_relu)
    relu_k<<<cdiv((long long)N * HC, 256), 256, 0, stream>>>(obuf, N * HC);
}

extern "C" void kernel_launch(void* const* d_in, const int* in_sizes, int n_in,
                              void* d_out, int out_size, void* d_ws, size_t ws_size,
                              hipStream_t stream) {
  (void)n_in; (void)out_size; (void)ws_size;

  const float* x     = (const float*)d_in[0];
  const int*   ei    = (const int*)d_in[1];
  const float* ea    = (const float*)d_in[2];
  const int*   batch = (const int*)d_in[3];
  const float* Wm[3]   = {(const float*)d_in[4],  (const float*)d_in[10], (const float*)d_in[16]};
  const float* Wem[3]  = {(const float*)d_in[5],  (const float*)d_in[11], (const float*)d_in[17]};
  const float* atts[3] = {(const float*)d_in[6],  (const float*)d_in[12], (const float*)d_in[18]};
  const float* attd[3] = {(const float*)d_in[7],  (const float*)d_in[13], (const float*)d_in[19]};
  const float* atte[3] = {(const float*)d_in[8],  (const float*)d_in[14], (const float*)d_in[20]};
  const float* bb[3]   = {(const float*)d_in[9],  (const float*)d_in[15], (const float*)d_in[21]};

  const int N  = in_sizes[0] / 128;
  const int E  = in_sizes[2] / 3;
  const int Et = E + N;
  const int* src = ei;
  const int* dst = ei + E;
  const float invE = 1.0f / (float)E;

  // workspace carve (floats)
  float* wsf = (float*)d_ws;
  size_t p = 0;
  auto alloc = [&](size_t n) { float* r = wsf + p; p += n; return r; };
  float*    bufA   = alloc((size_t)N * 128);
  float*    bufB   = alloc((size_t)N * 128);
  float*    h3     = alloc((size_t)N * 10);
  float*    out3   = alloc((size_t)N * 10);
  float*    as_    = alloc((size_t)N * 4);
  float*    ad_    = alloc((size_t)N * 4);
  float*    denom  = alloc((size_t)N * 4);
  unsigned* mkey   = (unsigned*)alloc((size_t)N * 4);
  float*    a_edge = alloc((size_t)Et * 4);
  float*    Qbuf   = alloc(16);
  float*    easum  = alloc(4);
  float*    pooled = alloc(64 * 10);
  float*    cnt    = alloc(64);

  // edge_attr mean (self-loop fill)
  zero_f32<<<1, 64, 0, stream>>>(easum, 4);
  ea_sum<<<256, 256, 0, stream>>>(ea, easum, E);

  run_layer<4, 32>(x,    Wm[0], Wem[0], atts[0], attd[0], atte[0], bb[0],
                   bufA, bufB, src, dst, ea, easum, invE, Qbuf, as_, ad_,
                   denom, mkey, a_edge, N, E, true, stream);
  run_layer<4, 32>(bufB, Wm[1], Wem[1], atts[1], attd[1], atte[1], bb[1],
                   bufA, bufB, src, dst, ea, easum, invE, Qbuf, as_, ad_,
                   denom, mkey, a_edge, N, E, true, stream);
  run_layer<1, 10>(bufB, Wm[2], Wem[2], atts[2], attd[2], atte[2], bb[2],
                   h3, out3, src, dst, ea, easum, invE, Qbuf, as_, ad_,
                   denom, mkey, a_edge, N, E, false, stream);

  // global mean pool + log_softmax
  zero_f32<<<cdiv(64 * 10, 64), 64, 0, stream>>>(pooled, 64 * 10);
  zero_f32<<<1, 64, 0, stream>>>(cnt, 64);
  pool_kernel<<<cdiv((long long)N * 10, 256), 256, 0, stream>>>(
      out3, batch, pooled, cnt, N, 10);
  finalize_k<<<1, 64, 0, stream>>>(pooled, cnt, (float*)d_out, 64, 10);
}